// MoE_8229157339845
// MI455X (gfx1250) — compile-verified
//
#include <hip/hip_runtime.h>
#include <hip/hip_bf16.h>
#include <math.h>

typedef __attribute__((ext_vector_type(2))) float v2f;
typedef __attribute__((ext_vector_type(8))) float v8f;
typedef __attribute__((ext_vector_type(4))) int   v4i_;

#define T_TOK   4096
#define DMODEL  1024
#define E_N     8
#define DFF_N   4096
#define NSLOT_CAP (T_TOK*2 + 16*E_N)   /* 8320: buckets padded to 16 */
#define MTILES  (NSLOT_CAP/16)         /* 520 */

#define XS_STRIDE 516   /* 512 + 4: stride%64==4 -> conflict-free wave-wide b64 A reads */
#define HS_STRIDE 132   /* 128 + 4 */

/* workspace byte offsets */
#define WS_COUNT   0
#define WS_PO      64
#define WS_PAIR_E  128
#define WS_PAIR_POS (WS_PAIR_E  + 8192*4)
#define WS_PAIR_W   (WS_PAIR_POS + 8192*4)
#define WS_BTOK     (WS_PAIR_W   + 8192*4)
#define WS_BW       (WS_BTOK + NSLOT_CAP*4)
#define WS_H        262144             /* H: NSLOT_CAP x DFF floats */

#if defined(__HIP_DEVICE_COMPILE__) && \
    __has_builtin(__builtin_amdgcn_global_load_async_to_lds_b128) && \
    __has_builtin(__builtin_amdgcn_s_wait_asynccnt)
#define HAS_ASYNC 1
#else
#define HAS_ASYNC 0
#endif

#if HAS_ASYNC
#define ASYNC_CP128(gp, lp) \
  __builtin_amdgcn_global_load_async_to_lds_b128( \
      (__attribute__((address_space(1))) v4i_*)(gp), \
      (__attribute__((address_space(3))) v4i_*)(lp), 0, 0)
#endif

__device__ __forceinline__ float fast_rcp(float v) {
#if defined(__HIP_DEVICE_COMPILE__) && __has_builtin(__builtin_amdgcn_rcpf)
  return __builtin_amdgcn_rcpf(v);
#else
  return 1.0f / v;
#endif
}

__global__ __launch_bounds__(256) void moe_zero(float* __restrict__ out, int out_n,
                                                int* __restrict__ cnt,
                                                int* __restrict__ btok,
                                                float* __restrict__ bw) {
  int idx = blockIdx.x * 256 + threadIdx.x;
  if (idx < out_n) out[idx] = 0.0f;
  if (idx < E_N)   cnt[idx] = 0;
  if (idx < NSLOT_CAP) { btok[idx] = -1; bw[idx] = 0.0f; }
}

__global__ __launch_bounds__(256) void moe_router(const float* __restrict__ x,
                                                  const float* __restrict__ Wg,
                                                  const float* __restrict__ bg,
                                                  int* __restrict__ cnt,
                                                  int* __restrict__ pair_e,
                                                  int* __restrict__ pair_pos,
                                                  float* __restrict__ pair_w) {
  int wave = threadIdx.x >> 5;
  int lane = threadIdx.x & 31;
  int t = blockIdx.x * 8 + wave;
  if (t >= T_TOK) return;
  const float* xr = x + (size_t)t * DMODEL;
  float p[E_N];
#pragma unroll
  for (int e = 0; e < E_N; ++e) p[e] = 0.0f;
  for (int k = lane; k < DMODEL; k += 32) {
    float xv = xr[k];
#pragma unroll
    for (int e = 0; e < E_N; ++e) p[e] += xv * Wg[e * DMODEL + k];
  }
#pragma unroll
  for (int e = 0; e < E_N; ++e)
    for (int m = 16; m >= 1; m >>= 1) p[e] += __shfl_xor(p[e], m, 32);
  if (lane == 0) {
    float lg[E_N];
#pragma unroll
    for (int e = 0; e < E_N; ++e) lg[e] = p[e] + bg[e];
    int i0 = 0; float v0 = lg[0];
#pragma unroll
    for (int e = 1; e < E_N; ++e) if (lg[e] > v0) { v0 = lg[e]; i0 = e; }
    int i1 = (i0 == 0) ? 1 : 0; float v1 = lg[i1];
#pragma unroll
    for (int e = 0; e < E_N; ++e)
      if (e != i0 && e != i1 && lg[e] > v1) { v1 = lg[e]; i1 = e; }
    float w0 = fast_rcp(1.0f + __expf(v1 - v0));   /* 2-way softmax */
    float w1 = 1.0f - w0;
    int pos0 = atomicAdd(&cnt[i0], 1);
    int pos1 = atomicAdd(&cnt[i1], 1);
    pair_e[t*2]   = i0; pair_pos[t*2]   = pos0; pair_w[t*2]   = w0;
    pair_e[t*2+1] = i1; pair_pos[t*2+1] = pos1; pair_w[t*2+1] = w1;
  }
}

__global__ void moe_prefix(const int* __restrict__ cnt, int* __restrict__ po) {
  if (threadIdx.x == 0 && blockIdx.x == 0) {
    int acc = 0;
    for (int e = 0; e < E_N; ++e) { po[e] = acc; acc += (cnt[e] + 15) & ~15; }
    po[E_N] = acc;
  }
}

__global__ __launch_bounds__(256) void moe_build(const int* __restrict__ pair_e,
                                                 const int* __restrict__ pair_pos,
                                                 const float* __restrict__ pair_w,
                                                 const int* __restrict__ po,
                                                 int* __restrict__ btok,
                                                 float* __restrict__ bw) {
  int i = blockIdx.x * 256 + threadIdx.x;
  if (i >= T_TOK * 2) return;
  int e = pair_e[i];
  int slot = po[e] + pair_pos[i];
  btok[slot] = i >> 1;
  bw[slot] = pair_w[i];
}

/* H[slot, f] = silu(x @ W1^T) * (x @ W3^T); block = 16 slots x 128 DFF cols */
__global__ __launch_bounds__(256) void moe_gemm1(const float* __restrict__ x,
                                                 const float* __restrict__ W1,
                                                 const float* __restrict__ W3,
                                                 const int* __restrict__ po,
                                                 const int* __restrict__ btok,
                                                 float* __restrict__ H) {
  __shared__ float Xs[16 * XS_STRIDE];   /* row-major K-half of X, 33KB */
  __shared__ int poS[E_N + 1];
  __shared__ int toks[16];
  int tid = threadIdx.x;
  if (tid < E_N + 1) poS[tid] = po[tid];
  __syncthreads();
  int slot0 = blockIdx.x * 16;
  if (slot0 >= poS[E_N]) return;            /* uniform exit */
  int e = 0;
  while (e < E_N - 1 && slot0 >= poS[e + 1]) ++e;
  if (tid < 16) toks[tid] = btok[slot0 + tid];
  __syncthreads();
  bool allv = (toks[15] >= 0);   /* pads are a suffix of each bucket */

  int wave = tid >> 5, lane = tid & 31;
  int ln = lane & 15;            /* A row m and B col n */
  int hi = lane >> 4;            /* K half within the 16x16x4 op */
  int fbase = blockIdx.y * 128 + wave * 16;
  const float* w1p = W1 + (size_t)e * DFF_N * DMODEL + (size_t)(fbase + ln) * DMODEL;
  const float* w3p = W3 + (size_t)e * DFF_N * DMODEL + (size_t)(fbase + ln) * DMODEL;

  v8f acc1 = {}; v8f acc3 = {};
  for (int kh = 0; kh < 2; ++kh) {
    __syncthreads();
#if HAS_ASYNC
    if (allv) {
      /* 512 async B128 copies: global rows -> LDS rows, ASYNCcnt-tracked */
#pragma unroll
      for (int i = 0; i < 2; ++i) {
        int lin = i * 256 + tid;         /* 0..511 */
        int m = lin >> 5, seg = lin & 31;
        const float* gp = x + (size_t)toks[m] * DMODEL + kh * 512 + seg * 4;
        float* lp = &Xs[m * XS_STRIDE + seg * 4];
        ASYNC_CP128(gp, lp);
      }
      __builtin_amdgcn_s_wait_asynccnt(0);
    } else
#endif
    {
      for (int i = 0; i < 32; ++i) {
        int lin = i * 256 + tid;
        int m = lin >> 9, kk = lin & 511;
        int t = toks[m];
        Xs[m * XS_STRIDE + kk] = (t >= 0) ? x[(size_t)t * DMODEL + kh * 512 + kk] : 0.0f;
      }
    }
    __syncthreads();
    int kg = kh * 512;
    for (int kb = 0; kb < 512; kb += 64) {
      __builtin_prefetch(w1p + kg + kb + 64, 0, 0);
      __builtin_prefetch(w3p + kg + kb + 64, 0, 0);
#pragma unroll
      for (int kk = kb; kk < kb + 64; kk += 4) {
        int ka = kk + 2 * hi;
        v2f a; a.x = Xs[ln * XS_STRIDE + ka]; a.y = Xs[ln * XS_STRIDE + ka + 1];
        v2f b1; b1.x = w1p[kg + ka]; b1.y = w1p[kg + ka + 1];
        v2f b3; b3.x = w3p[kg + ka]; b3.y = w3p[kg + ka + 1];
        acc1 = __builtin_amdgcn_wmma_f32_16x16x4_f32(false, a, false, b1, (short)0, acc1, false, false);
        acc3 = __builtin_amdgcn_wmma_f32_16x16x4_f32(false, a, false, b3, (short)0, acc3, false, false);
      }
    }
  }
#pragma unroll
  for (int r = 0; r < 8; ++r) {
    float g = acc1[r];
    float s = g * fast_rcp(1.0f + __expf(-g));      /* silu via v_rcp_f32 */
    float hv = s * acc3[r];
    int m = r + (hi ? 8 : 0);
    H[(size_t)(slot0 + m) * DFF_N + fbase + ln] = hv;
  }
}

/* out[t, :] += w * (H[slot, :] @ W2_e^T); block = 16 slots x 128 D cols */
__global__ __launch_bounds__(256) void moe_gemm2(const float* __restrict__ W2,
                                                 const int* __restrict__ po,
                                                 const int* __restrict__ btok,
                                                 const float* __restrict__ bw,
                                                 const float* __restrict__ H,
                                                 float* __restrict__ out) {
  __shared__ float Hs[16 * HS_STRIDE];  /* row-major K-chunk of H, 8.4KB */
  __shared__ int poS[E_N + 1];
  __shared__ int toks[16];
  __shared__ float wts[16];
  int tid = threadIdx.x;
  if (tid < E_N + 1) poS[tid] = po[tid];
  __syncthreads();
  int slot0 = blockIdx.x * 16;
  if (slot0 >= poS[E_N]) return;
  int e = 0;
  while (e < E_N - 1 && slot0 >= poS[e + 1]) ++e;
  if (tid < 16) { toks[tid] = btok[slot0 + tid]; wts[tid] = bw[slot0 + tid]; }
  __syncthreads();

  int wave = tid >> 5, lane = tid & 31;
  int ln = lane & 15;
  int hi = lane >> 4;
  int dbase = blockIdx.y * 128 + wave * 16;
  const float* w2p = W2 + (size_t)e * DMODEL * DFF_N + (size_t)(dbase + ln) * DFF_N;

  v8f acc = {};
  for (int k0 = 0; k0 < DFF_N; k0 += 128) {
    __syncthreads();
#if HAS_ASYNC
    {
      /* H rows exist for padded slots too (zeros) -> unconditional async staging */
#pragma unroll
      for (int i = 0; i < 2; ++i) {
        int lin = i * 256 + tid;         /* 0..511 */
        int m = lin >> 5, seg = lin & 31;
        const float* gp = H + (size_t)(slot0 + m) * DFF_N + k0 + seg * 4;
        float* lp = &Hs[m * HS_STRIDE + seg * 4];
        ASYNC_CP128(gp, lp);
      }
      __builtin_amdgcn_s_wait_asynccnt(0);
    }
#else
    for (int i = 0; i < 8; ++i) {
      int lin = i * 256 + tid;
      int m = lin >> 7, kk = lin & 127;
      Hs[m * HS_STRIDE + kk] = H[(size_t)(slot0 + m) * DFF_N + k0 + kk];
    }
#endif
    __syncthreads();
    for (int kb = 0; kb < 128; kb += 64) {
      __builtin_prefetch(w2p + k0 + kb + 128, 0, 0);
#pragma unroll
      for (int kk = kb; kk < kb + 64; kk += 4) {
        int ka = kk + 2 * hi;
        v2f a; a.x = Hs[ln * HS_STRIDE + ka]; a.y = Hs[ln * HS_STRIDE + ka + 1];
        v2f b; b.x = w2p[k0 + ka]; b.y = w2p[k0 + ka + 1];
        acc = __builtin_amdgcn_wmma_f32_16x16x4_f32(false, a, false, b, (short)0, acc, false, false);
      }
    }
  }
#pragma unroll
  for (int r = 0; r < 8; ++r) {
    int m = r + (hi ? 8 : 0);
    int t = toks[m];
    if (t >= 0)
      atomicAdd(&out[(size_t)t * DMODEL + dbase + ln], wts[m] * acc[r]);
  }
}

extern "C" void kernel_launch(void* const* d_in, const int* in_sizes, int n_in,
                              void* d_out, int out_size, void* d_ws, size_t ws_size,
                              hipStream_t stream) {
  (void)in_sizes; (void)n_in; (void)out_size; (void)ws_size;
  const float* x  = (const float*)d_in[0];
  const float* Wg = (const float*)d_in[1];
  const float* bg = (const float*)d_in[2];
  const float* W1 = (const float*)d_in[3];
  const float* W2 = (const float*)d_in[4];
  const float* W3 = (const float*)d_in[5];
  /* d_in[6] = top_k (==2), baked in at compile time */
  float* out = (float*)d_out;
  char*  ws  = (char*)d_ws;
  int*   cnt      = (int*)  (ws + WS_COUNT);
  int*   po       = (int*)  (ws + WS_PO);
  int*   pair_e   = (int*)  (ws + WS_PAIR_E);
  int*   pair_pos = (int*)  (ws + WS_PAIR_POS);
  float* pair_w   = (float*)(ws + WS_PAIR_W);
  int*   btok     = (int*)  (ws + WS_BTOK);
  float* bw       = (float*)(ws + WS_BW);
  float* H        = (float*)(ws + WS_H);

  int out_n = T_TOK * DMODEL;
  moe_zero  <<<(out_n + 255) / 256, 256, 0, stream>>>(out, out_n, cnt, btok, bw);
  moe_router<<<T_TOK / 8, 256, 0, stream>>>(x, Wg, bg, cnt, pair_e, pair_pos, pair_w);
  moe_prefix<<<1, 32, 0, stream>>>(cnt, po);
  moe_build <<<(T_TOK * 2 + 255) / 256, 256, 0, stream>>>(pair_e, pair_pos, pair_w, po, btok, bw);
  dim3 g1(MTILES, DFF_N / 128);
  moe_gemm1 <<<g1, 256, 0, stream>>>(x, W1, W3, po, btok, H);
  dim3 g2(MTILES, DMODEL / 128);
  moe_gemm2 <<<g2, 256, 0, stream>>>(W2, po, btok, bw, H, out);
}